// GELU130_39857296507240
// MI455X (gfx1250) — compile-verified
//
#include <hip/hip_runtime.h>
#include <hip/hip_bf16.h>

// x: (B=16, T=2048, D=1024) fp32. Sequential EMA scan over T per (b,d) column,
// fused with gated tanh-GELU + global-stat gate. Memory bound: 256 MiB traffic,
// ~11 us roofline at 23.3 TB/s. One thread per (b,d) column; async global->LDS
// quad-buffered staging (CDNA5 ASYNCcnt path), 3 chunks (24 KB/wave) in flight.

#define B_DIM 16
#define T_DIM 2048
#define D_DIM 1024
#define BLOCK 256                 // one block = 256 contiguous d of one b
#define CHUNK 64                  // time steps per LDS buffer
#define NBUF  4                   // 4 x 64 KiB LDS: 2 iterations of reuse slack
                                  // (async LDS writes are NOT ordered vs
                                  // same-wave ds reads)
#define NCHUNK (T_DIM / CHUNK)    // 32
#define WPC (CHUNK / 4)           // async b128 ops per chunk per wave (16)

#if defined(__HIP_DEVICE_COMPILE__) && defined(__gfx1250__) &&                 \
    __has_builtin(__builtin_amdgcn_global_load_async_to_lds_b128) &&           \
    __has_builtin(__builtin_amdgcn_s_wait_asynccnt)
#define USE_ASYNC_LDS 1
#else
#define USE_ASYNC_LDS 0
#endif

#define AS1 __attribute__((address_space(1)))
#define AS3 __attribute__((address_space(3)))

typedef int v4i_t __attribute__((vector_size(16)));   // 16-byte payload type

#if defined(__HIP_DEVICE_COMPILE__) && __has_builtin(__builtin_amdgcn_tanhf)
__device__ __forceinline__ float tanh_fast(float x) {
  return __builtin_amdgcn_tanhf(x);     // v_tanh_f32 (gfx1250)
}
#elif defined(__HIP_DEVICE_COMPILE__) && __has_builtin(__builtin_amdgcn_tanh_f32)
__device__ __forceinline__ float tanh_fast(float x) {
  return __builtin_amdgcn_tanh_f32(x);  // alt builtin spelling
}
#else
__device__ __forceinline__ float tanh_fast(float x) {
  float ax = __builtin_fabsf(x);
  float e  = __expf(-2.0f * ax);
  float r  = (1.0f - e) * __builtin_amdgcn_rcpf(1.0f + e);
  return __builtin_copysignf(r, x);
}
#endif

__device__ __forceinline__ float softplus_f(float v) {
  // cheap (v_exp/v_log based); executed once per thread
  return fmaxf(v, 0.0f) + __logf(1.0f + __expf(-fabsf(v)));
}

struct Params {
  float dl, omdl;          // sigmoid(logit_decay_l), 1 - dl
  float sigma_l, alpha_l;  // softplus of local params
  float sigma_g, alpha_g;  // softplus of global params
  float mg, inv_sg;        // per-d global mean, 1/(sqrt(var_g)+eps)
};

__device__ __forceinline__ float step_process(float xt, bool first,
                                              float& mean, float& sq,
                                              const Params& p) {
  float gate;
  if (first) {
    mean = xt;
    sq   = xt * xt;
    gate = 1.0f;
  } else {
    float var  = fmaxf(fmaf(-mean, mean, sq), 1e-4f);
    float dinv = __builtin_amdgcn_rcpf(__builtin_amdgcn_sqrtf(var) + 1e-5f);
    float z    = (xt - mean) * dinv;
    gate = fmaf(p.alpha_l, tanh_fast(p.sigma_l * fabsf(z)), 1.0f);
    // carry update (the only loop-carried dependency: 2 FMAs)
    mean = fmaf(p.dl, mean, p.omdl * xt);
    sq   = fmaf(p.dl, sq, p.omdl * (xt * xt));
  }
  // gelu(x * gate)
  float u     = xt * gate;
  float u2    = u * u;
  float inner = u * fmaf(0.7978845608028654f * 0.044715f, u2,
                         0.7978845608028654f);
  float gelu  = 0.5f * u * (1.0f + tanh_fast(inner));
  // global post-gate
  float zg = (xt - p.mg) * p.inv_sg;
  float pg = fmaf(p.alpha_g, tanh_fast(p.sigma_g * fabsf(zg)), 1.0f);
  return gelu * pg;
}

__global__ __launch_bounds__(BLOCK) void ema_gate_gelu_kernel(
    const float* __restrict__ x,
    const float* __restrict__ p_dec_l,
    const float* __restrict__ p_lsig_l,
    const float* __restrict__ p_lalp_l,
    const float* __restrict__ p_lsig_g,
    const float* __restrict__ p_lalp_g,
    const float* __restrict__ emu,
    const float* __restrict__ esq,
    float* __restrict__ out) {
  const int tid  = threadIdx.x;
  const int b    = blockIdx.x >> 2;  // D/BLOCK = 4 blocks per batch
  const int dblk = blockIdx.x & 3;
  const int d    = dblk * BLOCK + tid;

  Params p;
  {
    float v   = p_dec_l[0];
    p.dl      = 1.0f / (1.0f + __expf(-v));
    p.omdl    = 1.0f - p.dl;
    p.sigma_l = softplus_f(p_lsig_l[0]);
    p.alpha_l = softplus_f(p_lalp_l[0]);
    p.sigma_g = softplus_f(p_lsig_g[0]);
    p.alpha_g = softplus_f(p_lalp_g[0]);
    p.mg      = emu[d];
    float sg  = esq[d];
    float vg  = fmaxf(fmaf(-p.mg, p.mg, sg), 1e-4f);
    p.inv_sg  = __builtin_amdgcn_rcpf(__builtin_amdgcn_sqrtf(vg) + 1e-5f);
  }

  const size_t base = (size_t)b * T_DIM * D_DIM + (size_t)dblk * BLOCK;
  const float* __restrict__ xc = x + base + tid;   // consumer column
  float* __restrict__ oc       = out + base + tid;

  float mean = 0.0f, sq = 0.0f;

#if USE_ASYNC_LDS
  // Quad-buffered LDS tile: CHUNK time-rows x 256 columns (4 x 64 KiB).
  __shared__ float lbuf[NBUF][CHUNK][BLOCK];
  // Loader mapping chosen so wave w only loads columns [32w, 32w+32) that it
  // itself consumes -> per-wave ASYNCcnt waits only, NO workgroup barriers.
  // One b128 op: 32 lanes x 16B = 4 rows x 128B of this wave's column slice.
  const int lane = tid & 31;
  const int wv   = tid >> 5;
  const int rrow = lane >> 3;                       // row-in-group 0..3
  const int lcol = (wv << 5) + ((lane & 7) << 2);   // column of 16B piece
  const float* __restrict__ gl =
      x + (size_t)b * T_DIM * D_DIM + (size_t)rrow * D_DIM + dblk * BLOCK + lcol;

#define ISSUE_CHUNK(c)                                                         \
  {                                                                            \
    const float* gp = gl + (size_t)(c) * CHUNK * D_DIM;                        \
    float* lp       = &lbuf[(c) % NBUF][rrow][lcol];                           \
    _Pragma("unroll")                                                          \
    for (int i = 0; i < WPC; ++i)                                              \
      __builtin_amdgcn_global_load_async_to_lds_b128(                          \
          (AS1 v4i_t*)(gp + (size_t)i * 4 * D_DIM),                            \
          (AS3 v4i_t*)(lp + i * 4 * BLOCK), 0, 0);                             \
  }

  ISSUE_CHUNK(0);
  ISSUE_CHUNK(1);
  ISSUE_CHUNK(2);               // 48 ops in flight (< 63 ASYNCcnt capacity)

  // ---- chunk 0 peeled: contains the scan-init (t==0) step ----
  __builtin_amdgcn_s_wait_asynccnt(2 * WPC);   // chunk 0 done; {1,2} in flight
  {
    const float(*lrow)[BLOCK] = lbuf[0];
#pragma unroll 8
    for (int r = 0; r < CHUNK; ++r) {
      float xt = lrow[r][tid];
      float y  = step_process(xt, r == 0, mean, sq, p);
      __builtin_nontemporal_store(y, oc + (size_t)r * D_DIM);
    }
  }

  // ---- steady state: prefetch issued BEFORE the wait, branch-free body ----
  for (int c = 1; c < NCHUNK; ++c) {
    if (c + 2 < NCHUNK) ISSUE_CHUNK(c + 2);     // -> {c, c+1, c+2} in flight
    if (c + 2 < NCHUNK)
      __builtin_amdgcn_s_wait_asynccnt(2 * WPC);  // chunk c done
    else if (c + 1 < NCHUNK)
      __builtin_amdgcn_s_wait_asynccnt(WPC);
    else
      __builtin_amdgcn_s_wait_asynccnt(0);
    const float(*lrow)[BLOCK] = lbuf[c % NBUF];
    const int tbase = c * CHUNK;
    for (int r0 = 0; r0 < CHUNK; r0 += 8) {
      float v[8];
#pragma unroll
      for (int j = 0; j < 8; ++j) v[j] = lrow[r0 + j][tid];
#pragma unroll
      for (int j = 0; j < 8; ++j) {
        float y = step_process(v[j], false, mean, sq, p);
        __builtin_nontemporal_store(y, oc + (size_t)(tbase + r0 + j) * D_DIM);
      }
    }
  }
#undef ISSUE_CHUNK
#else
  // Fallback: register-pipelined strided walk (4 loads in flight + L2 prefetch).
  for (int t = 0; t < T_DIM; t += 4) {
    float a0 = xc[(size_t)(t + 0) * D_DIM];
    float a1 = xc[(size_t)(t + 1) * D_DIM];
    float a2 = xc[(size_t)(t + 2) * D_DIM];
    float a3 = xc[(size_t)(t + 3) * D_DIM];
    if (t + 16 < T_DIM)
      __builtin_prefetch(xc + (size_t)(t + 16) * D_DIM, 0, 0);
    float y0 = step_process(a0, t == 0, mean, sq, p);
    float y1 = step_process(a1, false, mean, sq, p);
    float y2 = step_process(a2, false, mean, sq, p);
    float y3 = step_process(a3, false, mean, sq, p);
    __builtin_nontemporal_store(y0, oc + (size_t)(t + 0) * D_DIM);
    __builtin_nontemporal_store(y1, oc + (size_t)(t + 1) * D_DIM);
    __builtin_nontemporal_store(y2, oc + (size_t)(t + 2) * D_DIM);
    __builtin_nontemporal_store(y3, oc + (size_t)(t + 3) * D_DIM);
  }
#endif
}

extern "C" void kernel_launch(void* const* d_in, const int* in_sizes, int n_in,
                              void* d_out, int out_size, void* d_ws,
                              size_t ws_size, hipStream_t stream) {
  (void)in_sizes; (void)n_in; (void)out_size; (void)d_ws; (void)ws_size;
  const float* x        = (const float*)d_in[0];
  const float* dec_l    = (const float*)d_in[1];
  const float* lsig_l   = (const float*)d_in[2];
  const float* lalp_l   = (const float*)d_in[3];
  // d_in[4] = logit_decay_g: unused by the reference output
  const float* lsig_g   = (const float*)d_in[5];
  const float* lalp_g   = (const float*)d_in[6];
  const float* ema_mean = (const float*)d_in[7];
  const float* ema_sq   = (const float*)d_in[8];
  float* out            = (float*)d_out;

  dim3 grid(B_DIM * (D_DIM / BLOCK));  // 64 blocks = 512 wave32
  ema_gate_gelu_kernel<<<grid, BLOCK, 0, stream>>>(
      x, dec_l, lsig_l, lalp_l, lsig_g, lalp_g, ema_mean, ema_sq, out);
}